// FullyConnectedSteerableGeometricProductLayer_11879879544498
// MI455X (gfx1250) — compile-verified
//
#include <hip/hip_runtime.h>
#include <hip/hip_bf16.h>
#include <math.h>

typedef __attribute__((ext_vector_type(2))) float v2f;
typedef __attribute__((ext_vector_type(4))) float v4f;
typedef __attribute__((ext_vector_type(8))) float v8f;

#define BATCH 8
#define SEQ   1024
#define DIM   8

// ---------------------------------------------------------------------------
// Prep: per (b,s) compute q = norm(x W_q + b_q e0), k = norm(x W_k + b_k e0),
// then Aq[j][k] = sum_i q[i] * C[i,j,k]  (Cayley of Cl(3,0)), store Aq and k.
// ---------------------------------------------------------------------------

__device__ __forceinline__ void mv_normalize(float* v, const float* a) {
    const int lo[4] = {0, 1, 4, 7};
    const int hi[4] = {1, 4, 7, 8};
#pragma unroll
    for (int g = 0; g < 4; ++g) {
        float ss = 0.f;
#pragma unroll
        for (int i = lo[g]; i < hi[g]; ++i) ss += v[i] * v[i];
        float nrm = sqrtf(ss);
        float sa  = 1.f / (1.f + expf(-a[g]));          // sigmoid(a_norm[s][g])
        float f   = sa * (nrm - 1.f) + 1.f + 1e-6f;     // (adjusted norm) + EPS
        float inv = 1.f / f;
#pragma unroll
        for (int i = lo[g]; i < hi[g]; ++i) v[i] *= inv;
    }
}

__global__ __launch_bounds__(256) void gp_prep(
    const float* __restrict__ x,
    const float* __restrict__ qw, const float* __restrict__ qb,
    const float* __restrict__ kw, const float* __restrict__ kb,
    const float* __restrict__ an,
    float* __restrict__ Aq,        // [b][s][j][k] : 8*1024*64 floats
    float* __restrict__ Kn)        // [b][t][k]    : 8*1024*8 floats
{
    int tid = blockIdx.x * blockDim.x + threadIdx.x;   // 0 .. 8191 = (b,s)
    if (tid >= BATCH * SEQ) return;
    int s = tid & (SEQ - 1);

    float xv[8];
    const float* xp = x + (size_t)tid * 8;
#pragma unroll
    for (int d = 0; d < 8; ++d) xv[d] = xp[d];

    float q[8], k[8];
#pragma unroll
    for (int i = 0; i < 8; ++i) {
        float aq = 0.f, ak = 0.f;
#pragma unroll
        for (int d = 0; d < 8; ++d) {
            aq += xv[d] * qw[d * 8 + i];   // einsum('bsd,di->bsi')
            ak += xv[d] * kw[d * 8 + i];
        }
        q[i] = aq; k[i] = ak;
    }
    q[0] += qb[0];  // bias embeds into grade-0 blade only
    k[0] += kb[0];

    mv_normalize(q, an + (size_t)s * 4);
    mv_normalize(k, an + (size_t)s * 4);

    // Cayley contraction: blade order [1,e1,e2,e3,e12,e13,e23,e123].
    // perm maps blade-index <-> bitmask (involution swapping 3<->4).
    const int perm[8] = {0, 1, 2, 4, 3, 5, 6, 7};
    float Av[8][8];
#pragma unroll
    for (int j = 0; j < 8; ++j)
#pragma unroll
        for (int kk = 0; kk < 8; ++kk) Av[j][kk] = 0.f;

#pragma unroll
    for (int i = 0; i < 8; ++i) {
        int am = perm[i];
#pragma unroll
        for (int kk = 0; kk < 8; ++kk) {
            int bm = perm[kk];
            int swaps = 0;
#pragma unroll
            for (int bit = 0; bit < 3; ++bit)
                if ((bm >> bit) & 1) swaps += __popc(am >> (bit + 1));
            int   rm = am ^ bm;                       // Euclidean metric: e_i e_i = +1
            float sg = (swaps & 1) ? -1.f : 1.f;
            Av[perm[rm]][kk] += sg * q[i];
        }
    }

    float* ap = Aq + (size_t)tid * 64;
#pragma unroll
    for (int j = 0; j < 8; ++j) {
        v4f lo4 = {Av[j][0], Av[j][1], Av[j][2], Av[j][3]};
        v4f hi4 = {Av[j][4], Av[j][5], Av[j][6], Av[j][7]};
        *(v4f*)(ap + j * 8)     = lo4;
        *(v4f*)(ap + j * 8 + 4) = hi4;
    }
    float* kp = Kn + (size_t)tid * 8;
    v4f klo = {k[0], k[1], k[2], k[3]};
    v4f khi = {k[4], k[5], k[6], k[7]};
    *(v4f*)(kp)     = klo;
    *(v4f*)(kp + 4) = khi;
}

// ---------------------------------------------------------------------------
// GEMM: out[b,s,t,j] = sum_k Aq[b,s,j,k] * Kn[b,t,k].
// One wave = one 16(s) x 16(t) tile for ALL 8 j -> 16 x v_wmma_f32_16x16x4_f32.
// Lane&15 = M (A) / N (B); lane>>4 selects K half {0,1}/{2,3} (and {4,5}/{6,7}).
// Stores: per row, each lane emits two float4 (j=0..3, j=4..7) -> each 32-lane
// store pair fills 512B contiguous per s-row (one s-row per half-wave): ideal HBM write combining.
// ---------------------------------------------------------------------------

__global__ __launch_bounds__(256) void gp_gemm(
    const float* __restrict__ Aq,
    const float* __restrict__ Kn,
    float* __restrict__ out)
{
    int wave = threadIdx.x >> 5;
    int lane = threadIdx.x & 31;
    int tile = blockIdx.x * 8 + wave;      // 0 .. 32767
    int tT   = tile & 63;
    int sT   = (tile >> 6) & 63;
    int b    = tile >> 12;
    int half = lane >> 4;
    int l    = lane & 15;

    // B operand: Kn rows, N = t, K from lane half
    const float* kp = Kn + ((size_t)b * SEQ + (size_t)(tT * 16 + l)) * 8;
    v2f b0 = *(const v2f*)(kp + half * 2);
    v2f b1 = *(const v2f*)(kp + 4 + half * 2);

    const float* ap = Aq + ((size_t)b * SEQ + (size_t)(sT * 16 + l)) * 64;

    v8f acc[8];
#pragma unroll
    for (int j = 0; j < 8; ++j) {
        v2f a0 = *(const v2f*)(ap + j * 8 + half * 2);
        v2f a1 = *(const v2f*)(ap + j * 8 + 4 + half * 2);
        v8f c = {0.f, 0.f, 0.f, 0.f, 0.f, 0.f, 0.f, 0.f};
        c = __builtin_amdgcn_wmma_f32_16x16x4_f32(false, a0, false, b0,
                                                  (short)0, c, false, false);
        c = __builtin_amdgcn_wmma_f32_16x16x4_f32(false, a1, false, b1,
                                                  (short)0, c, false, false);
        acc[j] = c;
    }

    // D layout: VGPR v holds M = v + 8*half, N = lane&15.
#pragma unroll
    for (int v = 0; v < 8; ++v) {
        size_t srow = (size_t)(sT * 16 + half * 8 + v);
        float* op = out + (((size_t)b * SEQ + srow) * SEQ + (size_t)(tT * 16 + l)) * 8;
        v4f lo4 = {acc[0][v], acc[1][v], acc[2][v], acc[3][v]};
        v4f hi4 = {acc[4][v], acc[5][v], acc[6][v], acc[7][v]};
        *(v4f*)(op)     = lo4;
        *(v4f*)(op + 4) = hi4;
    }
}

// ---------------------------------------------------------------------------

extern "C" void kernel_launch(void* const* d_in, const int* in_sizes, int n_in,
                              void* d_out, int out_size, void* d_ws, size_t ws_size,
                              hipStream_t stream) {
    const float* x  = (const float*)d_in[0];
    const float* qw = (const float*)d_in[1];
    const float* qb = (const float*)d_in[2];
    const float* kw = (const float*)d_in[3];
    const float* kb = (const float*)d_in[4];
    const float* an = (const float*)d_in[5];

    float* Aq = (float*)d_ws;                       // 8*1024*64 f32 = 2 MB
    float* Kn = Aq + (size_t)BATCH * SEQ * 64;      // 8*1024*8  f32 = 256 KB

    gp_prep<<<(BATCH * SEQ + 255) / 256, 256, 0, stream>>>(x, qw, qb, kw, kb, an, Aq, Kn);

    // 8 b * 64 s-tiles * 64 t-tiles = 32768 waves, 8 waves per block
    gp_gemm<<<32768 / 8, 256, 0, stream>>>(Aq, Kn, (float*)d_out);
}